// LogSpaceSelectiveCell_27066883900115
// MI455X (gfx1250) — compile-verified
//
#include <hip/hip_runtime.h>
#include <hip/hip_bf16.h>
#include <math.h>

// ---------------------------------------------------------------------------
// LogSpaceSelectiveCell for MI455X (gfx1250, wave32, WMMA)
//
//   1) proj GEMMs (x @ W^T + b) for W_alpha, W_x, W_delta  -> ws
//   2) scan kernel: 16384 independent elementwise recurrences over T=512
//   3) compete kernel: softmax over groups of 32 == one wave32 (shfl reduce)
//   4) out GEMM: h_lin @ W_out^T with fused compete*silu epilogue
//
// GEMMs use V_WMMA_F32_16X16X4_F32, 64x128 block tile, 32x64 wave tile.
// 128-thread workgroups (4 wave32): in CU mode only 2 waves/SIMD must
// co-reside, doubling the VGPR cap vs 256-thread blocks and eliminating
// the scratch spills of the register-staged double-buffer pipeline.
// Contiguous 8B LDS fragment reads (ds_load_2addr_b64, conflict-free
// stride 36); A-next / B-next committed at half-chunk boundaries.
// ---------------------------------------------------------------------------

typedef __attribute__((ext_vector_type(2))) float v2f;
typedef __attribute__((ext_vector_type(8))) float v8f;

static constexpr int kT = 512;
static constexpr int kB = 16;
static constexpr int kD = 1024;
static constexpr int kM = kT * kB;       // 8192 GEMM rows
static constexpr float kLogZeroEps = 1e-12f;

// GEMM tiling
static constexpr int BM = 64;    // block rows
static constexpr int BN = 128;   // block cols
static constexpr int KC = 32;    // K chunk staged in LDS
static constexpr int KS = KC + 4; // row stride (floats): 16B-aligned f4 stores,
                                  // conflict-free strided fragment reads
static constexpr int NTHREADS = 128;   // 4 wave32

__device__ __forceinline__ v8f wmma_f32(v2f a, v2f b, v8f c) {
    return __builtin_amdgcn_wmma_f32_16x16x4_f32(
        /*neg_a=*/false, a, /*neg_b=*/false, b,
        /*c_mod=*/(short)0, c, /*reuse_a=*/false, /*reuse_b=*/false);
}

// ---------------------------------------------------------------------------
// GEMM: C[m,n] = sum_k A[m,k] * W[n,k]   (A: [M,K], W: [N,K], row-major)
// EPI==0: C = acc + bias[n]
// EPI==1: C = compete[m,n] * silu(acc)
// ---------------------------------------------------------------------------
template <int EPI>
__global__ __launch_bounds__(NTHREADS)
__attribute__((amdgpu_waves_per_eu(1)))
void gemm_wmma_kernel(
    const float* __restrict__ A, const float* __restrict__ W,
    const float* __restrict__ bias, const float* __restrict__ compete,
    float* __restrict__ C, int M, int N, int K)
{
    __shared__ float As[2][BM][KS];
    __shared__ float Bs[2][BN][KS];

    const int tid  = threadIdx.x;
    const int lane = tid & 31;
    const int wave = tid >> 5;       // 4 waves: 2 (M) x 2 (N)
    const int wm   = wave >> 1;      // 0..1 -> 32-row strip
    const int wn   = wave & 1;       // 0..1 -> 64-col strip
    const int m0   = blockIdx.x * BM;
    const int n0   = blockIdx.y * BN;
    const int half = lane >> 4;      // 0: lanes 0-15 (k,k+1), 1: lanes 16-31 (k+2,k+3)
    const int l15  = lane & 15;

    // loader mapping (float4 granularity): 8 f4 per 32-float row
    const int lrow = tid >> 3;             // 0..15 base row (stride 16 per iter)
    const int lcol = (tid & 7) * 4;        // f4 column within 32-float row

    v8f acc[2][4];
    #pragma unroll
    for (int s = 0; s < 2; ++s)
        #pragma unroll
        for (int t = 0; t < 4; ++t) acc[s][t] = (v8f)0.0f;

    const int nch = K / KC;
    float4 ra[4], rb[8];

    // ---- prologue: chunk 0 -> buffer 0 ----
    #pragma unroll
    for (int it = 0; it < 4; ++it)
        ra[it] = *(const float4*)&A[(size_t)(m0 + lrow + it * 16) * K + lcol];
    #pragma unroll
    for (int it = 0; it < 8; ++it)
        rb[it] = *(const float4*)&W[(size_t)(n0 + lrow + it * 16) * K + lcol];
    #pragma unroll
    for (int it = 0; it < 4; ++it)
        *(float4*)&As[0][lrow + it * 16][lcol] = ra[it];
    #pragma unroll
    for (int it = 0; it < 8; ++it)
        *(float4*)&Bs[0][lrow + it * 16][lcol] = rb[it];
    __syncthreads();

    for (int ch = 0; ch < nch; ++ch) {
        const int cur = ch & 1;
        const int nxt = cur ^ 1;
        const int k0n = (ch + 1) * KC;
        const bool more = (ch + 1 < nch);

        // stage next A chunk (latency hidden by first half of WMMA stream)
        if (more) {
            #pragma unroll
            for (int it = 0; it < 4; ++it)
                ra[it] = *(const float4*)&A[(size_t)(m0 + lrow + it * 16) * K + k0n + lcol];
        }

        // ---- first half: 4 k-steps, 8 WMMAs each ----
        #pragma unroll
        for (int kk = 0; kk < KC / 2; kk += 4) {
            const int kb = kk + 2 * half;   // even -> 8B-aligned pair
            const v2f a0 = *(const v2f*)&As[cur][wm * 32 + l15][kb];
            const v2f a1 = *(const v2f*)&As[cur][wm * 32 + 16 + l15][kb];
            #pragma unroll
            for (int t = 0; t < 4; ++t) {
                const v2f b = *(const v2f*)&Bs[cur][wn * 64 + t * 16 + l15][kb];
                acc[0][t] = wmma_f32(a0, b, acc[0][t]);
                acc[1][t] = wmma_f32(a1, b, acc[1][t]);
            }
        }

        // commit A-next, stage B-next (hidden by second half)
        if (more) {
            #pragma unroll
            for (int it = 0; it < 4; ++it)
                *(float4*)&As[nxt][lrow + it * 16][lcol] = ra[it];
            #pragma unroll
            for (int it = 0; it < 8; ++it)
                rb[it] = *(const float4*)&W[(size_t)(n0 + lrow + it * 16) * K + k0n + lcol];
        }

        // ---- second half: 4 k-steps, 8 WMMAs each ----
        #pragma unroll
        for (int kk = KC / 2; kk < KC; kk += 4) {
            const int kb = kk + 2 * half;
            const v2f a0 = *(const v2f*)&As[cur][wm * 32 + l15][kb];
            const v2f a1 = *(const v2f*)&As[cur][wm * 32 + 16 + l15][kb];
            #pragma unroll
            for (int t = 0; t < 4; ++t) {
                const v2f b = *(const v2f*)&Bs[cur][wn * 64 + t * 16 + l15][kb];
                acc[0][t] = wmma_f32(a0, b, acc[0][t]);
                acc[1][t] = wmma_f32(a1, b, acc[1][t]);
            }
        }

        // commit B-next
        if (more) {
            #pragma unroll
            for (int it = 0; it < 8; ++it)
                *(float4*)&Bs[nxt][lrow + it * 16][lcol] = rb[it];
        }
        __syncthreads();
    }

    // ---- epilogue: D layout -> m_local = v + 8*half, n_local = lane&15 ----
    #pragma unroll
    for (int s = 0; s < 2; ++s) {
        #pragma unroll
        for (int t = 0; t < 4; ++t) {
            const int n = n0 + wn * 64 + t * 16 + l15;
            #pragma unroll
            for (int v = 0; v < 8; ++v) {
                const int m = m0 + wm * 32 + s * 16 + 8 * half + v;
                const size_t off = (size_t)m * N + n;
                const float val = acc[s][t][v];
                if (EPI == 0) {
                    C[off] = val + bias[n];
                } else {
                    const float sig = val / (1.0f + __expf(-val));   // silu
                    C[off] = compete[off] * sig;
                }
            }
        }
    }
}

// ---------------------------------------------------------------------------
// Scan: 16384 independent recurrences (one per (b,d)), loop over T=512.
// ---------------------------------------------------------------------------
__device__ __forceinline__ float softplus_f(float x) {
    return fmaxf(x, 0.0f) + log1pf(__expf(-fabsf(x)));
}

__device__ __forceinline__ void signed_log_add(
    float la, float sa, float lb, float sb, float& lo, float& so)
{
    float m = fmaxf(la, lb);
    float s = sa * __expf(la - m) + sb * __expf(lb - m);
    lo = m + __logf(fabsf(s) + kLogZeroEps);
    so = (s >= 0.0f) ? 1.0f : -1.0f;
}

__global__ __launch_bounds__(256) void scan_kernel(
    const float* __restrict__ xa, const float* __restrict__ xl,
    const float* __restrict__ xd,
    const float* __restrict__ log_h0, const float* __restrict__ sign_h0,
    const float* __restrict__ log_r_h, const float* __restrict__ sign_r_h,
    float* __restrict__ logh_out,   // [(T+1),B,D]
    float* __restrict__ signh_out,  // [(T+1),B,D]
    float* __restrict__ hlin_out)   // [T,B,D]
{
    const int idx = blockIdx.x * blockDim.x + threadIdx.x;   // 0..B*D-1
    const int d   = idx & (kD - 1);
    const int BD  = kB * kD;

    const float lrh = log_r_h[d];
    const float srh = sign_r_h[d];
    float log_h  = log_h0[idx];
    float sign_h = sign_h0[idx];

    logh_out[idx]  = log_h;    // t = 0 slot
    signh_out[idx] = sign_h;

    for (int t = 0; t < kT; ++t) {
        const size_t off = (size_t)t * BD + idx;
        const float a_raw = xa[off];
        const float lin   = xl[off];
        const float d_raw = xd[off];

        const float alpha = 1.0f + softplus_f(a_raw);

        const float log_rh  = lrh + log_h;
        const float sign_rh = srh * sign_h;
        const float log_in  = __logf(fabsf(lin) + kLogZeroEps);
        const float sign_in = (lin >= 0.0f) ? 1.0f : -1.0f;

        float log_v, sign_v;
        signed_log_add(log_rh, sign_rh, log_in, sign_in, log_v, sign_v);

        const float log_cand = 10.0f * tanhf(alpha * log_v * 0.1f);  // soft_bound
        const float log_1md  = -softplus_f(d_raw);    // log_sigmoid(-d_raw)
        const float log_d    = -softplus_f(-d_raw);   // log_sigmoid(d_raw)

        float nl, ns;
        signed_log_add(log_1md + log_h, sign_h, log_d + log_cand, sign_v, nl, ns);
        log_h = nl;
        sign_h = ns;

        const float h_lin = sign_h * __expf(log_h);
        logh_out[(size_t)(t + 1) * BD + idx]  = log_h;
        signh_out[(size_t)(t + 1) * BD + idx] = sign_h;
        hlin_out[off] = h_lin;
    }
}

// ---------------------------------------------------------------------------
// Compete: softmax over contiguous groups of 32 == exactly one wave32.
// ---------------------------------------------------------------------------
__global__ __launch_bounds__(256) void compete_kernel(
    const float* __restrict__ hlin, float* __restrict__ comp)
{
    const size_t idx = (size_t)blockIdx.x * blockDim.x + threadIdx.x;
    float v = hlin[idx];
    float mx = v;
    #pragma unroll
    for (int m = 16; m >= 1; m >>= 1) mx = fmaxf(mx, __shfl_xor(mx, m, 32));
    float e = __expf(v - mx);
    float sum = e;
    #pragma unroll
    for (int m = 16; m >= 1; m >>= 1) sum += __shfl_xor(sum, m, 32);
    comp[idx] = e / sum;
}

// ---------------------------------------------------------------------------
extern "C" void kernel_launch(void* const* d_in, const int* in_sizes, int n_in,
                              void* d_out, int out_size, void* d_ws, size_t ws_size,
                              hipStream_t stream)
{
    (void)in_sizes; (void)n_in; (void)out_size; (void)ws_size;

    const float* x        = (const float*)d_in[0];
    const float* log_h0   = (const float*)d_in[1];
    const float* sign_h0  = (const float*)d_in[2];
    const float* W_x      = (const float*)d_in[3];
    const float* b        = (const float*)d_in[4];
    const float* log_r_h  = (const float*)d_in[5];
    const float* sign_r_h = (const float*)d_in[6];
    const float* W_alpha  = (const float*)d_in[7];
    const float* b_alpha  = (const float*)d_in[8];
    const float* W_delta  = (const float*)d_in[9];
    const float* b_delta  = (const float*)d_in[10];
    const float* W_out    = (const float*)d_in[11];
    // d_in[12] = n_groups (32, baked in as constant)

    const size_t TBD  = (size_t)kT * kB * kD;        // 8388608
    const size_t T1BD = (size_t)(kT + 1) * kB * kD;  // 8404992

    float* out        = (float*)d_out;
    float* out_output = out;                  // [T,B,D]
    float* out_logh   = out + TBD;            // [T+1,B,D]
    float* out_signh  = out_logh + T1BD;      // [T+1,B,D]
    float* out_hlin   = out_signh + T1BD;     // [T,B,D]

    float* ws   = (float*)d_ws;
    float* xa   = ws;                         // alpha_raw
    float* xl   = ws + TBD;                   // linear_input
    float* xd   = ws + 2 * TBD;               // delta_raw
    float* comp = ws + 3 * TBD;               // compete softmax

    dim3 gemm_grid(kM / BM, kD / BN);         // 128 x 8 blocks
    dim3 gemm_block(NTHREADS);

    // 1) input projections
    gemm_wmma_kernel<0><<<gemm_grid, gemm_block, 0, stream>>>(
        x, W_alpha, b_alpha, nullptr, xa, kM, kD, kD);
    gemm_wmma_kernel<0><<<gemm_grid, gemm_block, 0, stream>>>(
        x, W_x, b, nullptr, xl, kM, kD, kD);
    gemm_wmma_kernel<0><<<gemm_grid, gemm_block, 0, stream>>>(
        x, W_delta, b_delta, nullptr, xd, kM, kD, kD);

    // 2) elementwise scan (fully parallel across B*D)
    scan_kernel<<<(kB * kD) / 256, 256, 0, stream>>>(
        xa, xl, xd, log_h0, sign_h0, log_r_h, sign_r_h,
        out_logh, out_signh, out_hlin);

    // 3) group softmax (one wave32 per group of 32)
    compete_kernel<<<(unsigned)(TBD / 256), 256, 0, stream>>>(out_hlin, comp);

    // 4) output projection with fused compete * silu epilogue
    gemm_wmma_kernel<1><<<gemm_grid, gemm_block, 0, stream>>>(
        out_hlin, W_out, nullptr, comp, out_output, kM, kD, kD);
}